// SimpleGraphNeuralNet_19782619365957
// MI455X (gfx1250) — compile-verified
//
#include <hip/hip_runtime.h>

typedef float v2f __attribute__((ext_vector_type(2)));
typedef float v8f __attribute__((ext_vector_type(8)));

#define N_NODES 50000
#define N_EDGES 800000

__device__ __forceinline__ v8f wmma_f32(v2f a, v2f b, v8f c) {
    return __builtin_amdgcn_wmma_f32_16x16x4_f32(false, a, false, b, (short)0, c, false, false);
}

// ---------------- zero workspace (agg + stats) ----------------
__global__ void __launch_bounds__(256) zero_kernel(float* __restrict__ p, int n) {
    int i = blockIdx.x * 256 + threadIdx.x;
    if (i < n) p[i] = 0.0f;
}

// ---------------- pack weights: W[l][k][c] -> Wp[l][k/2][c] = (W[k][c], W[k+1][c]) ----------------
// Makes each WMMA B-fragment a single coalesced b64 load.
__global__ void __launch_bounds__(256)
pack_kernel(const float* __restrict__ W, float2* __restrict__ Wp,
            int K, int N, int nPairsTotal) {
    int i = blockIdx.x * 256 + threadIdx.x;
    if (i >= nPairsTotal) return;
    int k2N = (K >> 1) * N;
    int l = i / k2N, r = i - l * k2N;
    int k2 = r / N, col = r - k2 * N;
    const float* s = W + (size_t)l * K * N + (size_t)(2 * k2) * N + col;
    float2 v; v.x = s[0]; v.y = s[N];
    Wp[i] = v;
}

// ---------------- edge scatter-add: one wave per edge ----------------
__global__ void __launch_bounds__(256)
scatter_kernel(const float* __restrict__ x, const int* __restrict__ ei,
               float* __restrict__ agg) {
    int gwave = (blockIdx.x * 256 + threadIdx.x) >> 5;  // edge id (wave-uniform)
    int lane  = threadIdx.x & 31;
    if (gwave >= N_EDGES) return;
    int src = ei[gwave];
    int dst = ei[N_EDGES + gwave];
    float4 v = ((const float4*)(x + (size_t)src * 128))[lane];
    float* ad = agg + (size_t)dst * 128 + lane * 4;
    atomicAdd(ad + 0, v.x);
    atomicAdd(ad + 1, v.y);
    atomicAdd(ad + 2, v.z);
    atomicAdd(ad + 3, v.w);
}

// ---------------- GEMM1: t = ((1+eps)*x + agg) @ W1 + b1, accumulate BN stats ----------------
// grid = 1563 (32 rows each), 8 waves, wave w -> cols [w*32, w*32+32), 2 M-subtiles
__global__ void __launch_bounds__(256)
gemm1_bn_kernel(const float* __restrict__ x, const float* __restrict__ agg,
                const float2* __restrict__ W1pk, const float* __restrict__ b1p,
                const float* __restrict__ epsArr, int layer,
                float* __restrict__ t, float* __restrict__ stats) {
    __shared__ float As[32 * 132];          // stride 132: 132 % 64 = 4 -> conflict free
    const int mBase = blockIdx.x * 32;
    const int tid = threadIdx.x;
    const float epl = 1.0f + epsArr[layer];
#pragma unroll
    for (int i = 0; i < 16; ++i) {          // 32x128 A tile, fused h = (1+eps)x + agg
        int e = tid + i * 256;
        int r = e >> 7, k = e & 127;
        int gr = mBase + r; if (gr > N_NODES - 1) gr = N_NODES - 1;   // tail clamp
        int g = gr * 128 + k;
        As[r * 132 + k] = epl * x[g] + agg[g];
    }
    __syncthreads();

    const int wave = tid >> 5, lane = tid & 31;
    const int half = lane >> 4, l16 = lane & 15;
    const int c0 = wave * 32;
    const int col0 = c0 + l16, col1 = col0 + 16;
    const float2* Bl = W1pk + (size_t)layer * 64 * 256;
    v8f acc00 = {}, acc01 = {}, acc10 = {}, acc11 = {};
    for (int k = 0; k < 128; k += 4) {
        int kk = k + half * 2;              // A frag: v0=K0|K2, v1=K1|K3
        v2f a0; a0.x = As[l16 * 132 + kk];        a0.y = As[l16 * 132 + kk + 1];
        v2f a1; a1.x = As[(16 + l16) * 132 + kk]; a1.y = As[(16 + l16) * 132 + kk + 1];
        float2 p0 = Bl[((k >> 1) + half) * 256 + col0];   // single b64, coalesced
        float2 p1 = Bl[((k >> 1) + half) * 256 + col1];
        v2f b0; b0.x = p0.x; b0.y = p0.y;
        v2f b1; b1.x = p1.x; b1.y = p1.y;
        acc00 = wmma_f32(a0, b0, acc00);
        acc01 = wmma_f32(a0, b1, acc01);
        acc10 = wmma_f32(a1, b0, acc10);
        acc11 = wmma_f32(a1, b1, acc11);
    }
    const float* bb = b1p + layer * 256;
    float bias0 = bb[col0], bias1 = bb[col1];
#pragma unroll
    for (int m = 0; m < 2; ++m) {
        float s0 = 0.f, q0 = 0.f, s1 = 0.f, q1 = 0.f;
#pragma unroll
        for (int j = 0; j < 8; ++j) {       // D: vgpr j = row j (lanes 0-15) / j+8 (16-31)
            int row = mBase + m * 16 + j + half * 8;
            if (row < N_NODES) {
                float v0 = (m ? acc10[j] : acc00[j]) + bias0;
                float v1 = (m ? acc11[j] : acc01[j]) + bias1;
                t[row * 256 + col0] = v0;
                t[row * 256 + col1] = v1;
                s0 += v0; q0 += v0 * v0;
                s1 += v1; q1 += v1 * v1;
            }
        }
        s0 += __shfl_xor(s0, 16, 32); q0 += __shfl_xor(q0, 16, 32);
        s1 += __shfl_xor(s1, 16, 32); q1 += __shfl_xor(q1, 16, 32);
        if (half == 0) {
            atomicAdd(&stats[col0], s0); atomicAdd(&stats[256 + col0], q0);
            atomicAdd(&stats[col1], s1); atomicAdd(&stats[256 + col1], q1);
        }
    }
}

// ---------------- BN finalize: fold gamma/beta into per-column scale/shift ----------------
__global__ void __launch_bounds__(256)
bn_finalize_kernel(const float* __restrict__ gamma, const float* __restrict__ beta,
                   int layer, float* __restrict__ stats) {
    int c = threadIdx.x;
    float inv = 1.0f / (float)N_NODES;
    float mu  = stats[c] * inv;
    float var = stats[256 + c] * inv - mu * mu;
    float sc  = gamma[layer * 256 + c] * rsqrtf(var + 1e-5f);
    float sh  = beta[layer * 256 + c] - mu * sc;
    stats[c] = sc; stats[256 + c] = sh;
}

// ---------------- GEMM2: xout = relu(BN(t)) @ W2 + b2 ----------------
// grid = 1563 (32 rows), 8 waves, wave w -> cols [w*16, w*16+16), 2 M-subtiles
__global__ void __launch_bounds__(256)
gemm2_kernel(const float* __restrict__ t, const float* __restrict__ stats,
             const float2* __restrict__ W2pk, const float* __restrict__ b2p,
             int layer, float* __restrict__ xout) {
    __shared__ float As[32 * 260];          // 260 % 64 = 4 -> conflict free
    const int mBase = blockIdx.x * 32;
    const int tid = threadIdx.x;
#pragma unroll
    for (int i = 0; i < 32; ++i) {          // 32x256 tile, fused BN + ReLU
        int e = tid + i * 256;
        int r = e >> 8, k = e & 255;
        int gr = mBase + r; if (gr > N_NODES - 1) gr = N_NODES - 1;   // tail clamp
        float v = t[gr * 256 + k];
        v = v * stats[k] + stats[256 + k];
        As[r * 260 + k] = fmaxf(v, 0.0f);
    }
    __syncthreads();

    const int wave = tid >> 5, lane = tid & 31;
    const int half = lane >> 4, l16 = lane & 15;
    const int col = wave * 16 + l16;
    const float2* Bl = W2pk + (size_t)layer * 128 * 128;
    v8f acc0 = {}, acc1 = {};
    for (int k = 0; k < 256; k += 4) {
        int kk = k + half * 2;
        v2f a0; a0.x = As[l16 * 260 + kk];        a0.y = As[l16 * 260 + kk + 1];
        v2f a1; a1.x = As[(16 + l16) * 260 + kk]; a1.y = As[(16 + l16) * 260 + kk + 1];
        float2 p = Bl[((k >> 1) + half) * 128 + col];
        v2f b; b.x = p.x; b.y = p.y;
        acc0 = wmma_f32(a0, b, acc0);
        acc1 = wmma_f32(a1, b, acc1);
    }
    float bias = b2p[layer * 128 + col];
#pragma unroll
    for (int m = 0; m < 2; ++m) {
#pragma unroll
        for (int j = 0; j < 8; ++j) {
            int row = mBase + m * 16 + j + half * 8;
            if (row < N_NODES)
                xout[row * 128 + col] = (m ? acc1[j] : acc0[j]) + bias;
        }
    }
}

// ---------------- Final GEMM: out = x @ lin_W + lin_b (N=64) ----------------
// grid = 1563 (32 rows), waves 0-3 -> rows[0,16), waves 4-7 -> rows[16,32)
__global__ void __launch_bounds__(256)
gemm3_kernel(const float* __restrict__ x, const float2* __restrict__ linWpk,
             const float* __restrict__ linb, float* __restrict__ out) {
    __shared__ float As[32 * 132];
    const int mBase = blockIdx.x * 32;
    const int tid = threadIdx.x;
#pragma unroll
    for (int i = 0; i < 16; ++i) {          // 32x128 tile (rows clamped for tail)
        int e = tid + i * 256;
        int r = e >> 7, k = e & 127;
        int gr = mBase + r; if (gr > N_NODES - 1) gr = N_NODES - 1;
        As[r * 132 + k] = x[gr * 128 + k];
    }
    __syncthreads();

    const int wave = tid >> 5, lane = tid & 31;
    const int half = lane >> 4, l16 = lane & 15;
    const int mSub = wave >> 2;
    const int col = (wave & 3) * 16 + l16;
    const int arow = mSub * 16 + l16;
    v8f acc = {};
    for (int k = 0; k < 128; k += 4) {
        int kk = k + half * 2;
        v2f a; a.x = As[arow * 132 + kk]; a.y = As[arow * 132 + kk + 1];
        float2 p = linWpk[((k >> 1) + half) * 64 + col];
        v2f b; b.x = p.x; b.y = p.y;
        acc = wmma_f32(a, b, acc);
    }
    float bias = linb[col];
#pragma unroll
    for (int j = 0; j < 8; ++j) {
        int row = mBase + mSub * 16 + j + half * 8;
        if (row < N_NODES) out[row * 64 + col] = acc[j] + bias;
    }
}

extern "C" void kernel_launch(void* const* d_in, const int* in_sizes, int n_in,
                              void* d_out, int out_size, void* d_ws, size_t ws_size,
                              hipStream_t stream) {
    const float* x     = (const float*)d_in[0];
    const int*   ei    = (const int*)  d_in[1];
    const float* W1    = (const float*)d_in[2];
    const float* b1    = (const float*)d_in[3];
    const float* gamma = (const float*)d_in[4];
    const float* beta  = (const float*)d_in[5];
    const float* W2    = (const float*)d_in[6];
    const float* b2    = (const float*)d_in[7];
    const float* eps   = (const float*)d_in[8];
    const float* linW  = (const float*)d_in[9];
    const float* linb  = (const float*)d_in[10];
    float* out = (float*)d_out;

    // Workspace layout (floats):
    // agg[6.4M] | stats[512] | t[12.8M] | xA[6.4M] | xB[6.4M] | W1pk[131072] | W2pk[131072] | linWpk[8192]
    float*  ws     = (float*)d_ws;
    float*  agg    = ws;
    float*  stats  = ws + 6400000;
    float*  t      = ws + 6400512;
    float*  xA     = ws + 19200512;
    float*  xB     = ws + 25600512;
    float2* W1pk   = (float2*)(ws + 32000512);
    float2* W2pk   = (float2*)(ws + 32131584);
    float2* linWpk = (float2*)(ws + 32262656);

    // One-time (per launch) weight packing for b64 B-fragment loads
    pack_kernel<<<(65536 + 255) / 256, 256, 0, stream>>>(W1, W1pk, 128, 256, 65536);
    pack_kernel<<<(65536 + 255) / 256, 256, 0, stream>>>(W2, W2pk, 256, 128, 65536);
    pack_kernel<<<(4096 + 255) / 256, 256, 0, stream>>>(linW, linWpk, 128, 64, 4096);

    const int mBlocks = (N_NODES + 31) / 32;   // 1563
    const float* cur = x;
    for (int l = 0; l < 4; ++l) {
        float* nxt = (l & 1) ? xB : xA;
        zero_kernel<<<(6400512 + 255) / 256, 256, 0, stream>>>(agg, 6400512); // agg + stats
        scatter_kernel<<<N_EDGES / 8, 256, 0, stream>>>(cur, ei, agg);
        gemm1_bn_kernel<<<mBlocks, 256, 0, stream>>>(cur, agg, W1pk, b1, eps, l, t, stats);
        bn_finalize_kernel<<<1, 256, 0, stream>>>(gamma, beta, l, stats);
        gemm2_kernel<<<mBlocks, 256, 0, stream>>>(t, stats, W2pk, b2, l, nxt);
        cur = nxt;
    }
    gemm3_kernel<<<mBlocks, 256, 0, stream>>>(cur, linWpk, linb, out);
}